// Net_18081812316551
// MI455X (gfx1250) — compile-verified
//
#include <hip/hip_runtime.h>

typedef __attribute__((ext_vector_type(2))) float v2f;
typedef __attribute__((ext_vector_type(8))) float v8f;

#define GCN_N 100000
#define GCN_E 6400000
#define FIN 9
#define FMID 16

// ---------------- degree / normalization ----------------
__global__ void k_init_deg(float* __restrict__ deg, int n) {
    int i = blockIdx.x * blockDim.x + threadIdx.x;
    if (i < n) deg[i] = 1.0f;                       // self-loop contribution
}

__global__ void k_count_deg(const long long* __restrict__ col, float* __restrict__ deg, int e) {
    int i = blockIdx.x * blockDim.x + threadIdx.x;
    if (i < e) atomicAdd(&deg[(int)col[i]], 1.0f);
}

__global__ void k_deg_to_dinv(float* __restrict__ deg, int n) {
    int i = blockIdx.x * blockDim.x + threadIdx.x;
    if (i < n) {
        float d = deg[i];
        deg[i] = (d > 0.0f) ? rsqrtf(d) : 0.0f;     // in-place: deg -> dinv
    }
}

// ---------------- dense GEMM h1 = x @ W1 via V_WMMA_F32_16X16X4_F32 ----------------
// One wave32 per 16-row tile; N is an exact multiple of 16 (100000 = 6250*16).
// K = 9 padded to 12 -> exactly 3 chained 16x16x4 WMMAs, no dead zero-chunk.
// Loads are branchless: clamped in-bounds addresses + v_cndmask zero-select.
__global__ __launch_bounds__(256) void k_gemm1_wmma(const float* __restrict__ x,
                                                    const float* __restrict__ W1,
                                                    float* __restrict__ h1, int n) {
    const int lane = threadIdx.x & 31;
    const int wave = threadIdx.x >> 5;
    const int tile = blockIdx.x * 8 + wave;          // 8 waves / block
    const int m0 = tile * 16;
    if (m0 >= n) return;                             // wave-uniform: EXEC all-1 past here

    const int half  = lane >> 4;                     // 0: lanes 0-15, 1: lanes 16-31
    const int l16   = lane & 15;
    const int kbase = half * 2;                      // per-chunk K of the .x element

    // ---- B (wave-uniform, hoisted): chunk kc covers K = kc*4 + {kbase, kbase+1}
    v2f b0, b1, b2;
    b0.x = W1[(kbase + 0) * FMID + l16];
    b0.y = W1[(kbase + 1) * FMID + l16];
    b1.x = W1[(kbase + 4) * FMID + l16];
    b1.y = W1[(kbase + 5) * FMID + l16];
    // chunk 2: K = 8+kbase (valid only for half==0); K = 9+kbase never valid
    const int  k2   = 8 + kbase;                     // 8 or 10
    const bool k2ok = (k2 < FIN);                    // true only for lanes 0-15
    const int  k2c  = k2ok ? k2 : (FIN - 1);         // clamp -> in-bounds load
    {
        float t = W1[k2c * FMID + l16];
        b2.x = k2ok ? t : 0.0f;                      // cndmask, no branch
        b2.y = 0.0f;
    }

    // ---- A: row m = m0 + l16 (always valid), same K striping as B
    const float* __restrict__ xr = x + (size_t)(m0 + l16) * FIN;
    v2f a0, a1, a2;
    a0.x = xr[kbase + 0];
    a0.y = xr[kbase + 1];
    a1.x = xr[kbase + 4];
    a1.y = xr[kbase + 5];
    {
        float t = xr[k2c];
        a2.x = k2ok ? t : 0.0f;
        a2.y = 0.0f;
    }

    v8f c = {};
    c = __builtin_amdgcn_wmma_f32_16x16x4_f32(false, a0, false, b0, (short)0, c, false, false);
    c = __builtin_amdgcn_wmma_f32_16x16x4_f32(false, a1, false, b1, (short)0, c, false, false);
    c = __builtin_amdgcn_wmma_f32_16x16x4_f32(false, a2, false, b2, (short)0, c, false, false);

    // D layout: VGPR v -> row (m0 + v + half*8), col l16
    float* __restrict__ hrow = h1 + (size_t)(m0 + half * 8) * FMID + l16;
#pragma unroll
    for (int v = 0; v < 8; ++v) {
        hrow[v * FMID] = c[v];
    }
}

// ---------------- layer-1 scatter ----------------
__global__ void k_init_acc1(const float* __restrict__ h1, const float* __restrict__ dinv,
                            float* __restrict__ acc1, int n) {
    int t = blockIdx.x * blockDim.x + threadIdx.x;
    if (t < n * FMID) {
        int i = t >> 4;
        float d = dinv[i];
        acc1[t] = h1[t] * d * d;                     // self-loop term, no atomics needed
    }
}

__global__ void k_scatter1(const long long* __restrict__ row, const long long* __restrict__ col,
                           const float* __restrict__ dinv, const float* __restrict__ h1,
                           float* __restrict__ acc1, int e) {
    long long t = (long long)blockIdx.x * blockDim.x + threadIdx.x;
    if (t >= (long long)e * FMID) return;
    int ei = (int)(t >> 4);
    int f  = (int)(t & 15);
    int r = (int)row[ei];
    int c = (int)col[ei];
    float w = dinv[r] * dinv[c];
    atomicAdd(&acc1[c * FMID + f], h1[r * FMID + f] * w);
}

// ---------------- relu + bias + dot(W2) + self-loop init of layer-2 accumulator ----------------
__global__ void k_relu_dot(const float* __restrict__ acc1, const float* __restrict__ b1,
                           const float* __restrict__ W2, const float* __restrict__ dinv,
                           float* __restrict__ tnode, float* __restrict__ out_acc, int n) {
    int i = blockIdx.x * blockDim.x + threadIdx.x;
    if (i >= n) return;
    float s = 0.0f;
#pragma unroll
    for (int f = 0; f < FMID; ++f) {
        float h = acc1[i * FMID + f] + b1[f];
        h = h > 0.0f ? h : 0.0f;
        s += h * W2[f];
    }
    tnode[i] = s;
    float d = dinv[i];
    out_acc[i] = s * d * d;                          // self-loop term for layer 2
}

// ---------------- layer-2 scatter + finalize ----------------
__global__ void k_scatter2(const long long* __restrict__ row, const long long* __restrict__ col,
                           const float* __restrict__ dinv, const float* __restrict__ tnode,
                           float* __restrict__ out_acc, int e) {
    int ei = blockIdx.x * blockDim.x + threadIdx.x;
    if (ei >= e) return;
    int r = (int)row[ei];
    int c = (int)col[ei];
    atomicAdd(&out_acc[c], tnode[r] * dinv[r] * dinv[c]);
}

__global__ void k_finalize(const float* __restrict__ out_acc, const float* __restrict__ b2,
                           float* __restrict__ out, int n) {
    int i = blockIdx.x * blockDim.x + threadIdx.x;
    if (i < n) {
        float v = out_acc[i] + b2[0];
        out[i] = v > 0.0f ? v : 0.0f;
    }
}

extern "C" void kernel_launch(void* const* d_in, const int* in_sizes, int n_in,
                              void* d_out, int out_size, void* d_ws, size_t ws_size,
                              hipStream_t stream) {
    const float*     x    = (const float*)d_in[0];        // [N, 9]
    const long long* eidx = (const long long*)d_in[1];    // [2, E] int64
    const float*     W1   = (const float*)d_in[2];        // [9, 16]
    const float*     b1   = (const float*)d_in[3];        // [16]
    const float*     W2   = (const float*)d_in[4];        // [16, 1]
    const float*     b2   = (const float*)d_in[5];        // [1]
    float*           out  = (float*)d_out;                // [N] (N x 1)

    const int N = GCN_N;
    const int E = GCN_E;
    const long long* row = eidx;        // edge_index[0] = source
    const long long* col = eidx + E;    // edge_index[1] = destination

    // workspace layout (floats): dinv[N] | h1[16N] | acc1[16N] | t[N] | out_acc[N]
    float* dinv    = (float*)d_ws;
    float* h1      = dinv + N;
    float* acc1    = h1 + (size_t)N * FMID;
    float* tnode   = acc1 + (size_t)N * FMID;
    float* out_acc = tnode + N;

    const int B = 256;
    // 1) degree -> dinv
    k_init_deg   <<<(N + B - 1) / B, B, 0, stream>>>(dinv, N);
    k_count_deg  <<<(E + B - 1) / B, B, 0, stream>>>(col, dinv, E);
    k_deg_to_dinv<<<(N + B - 1) / B, B, 0, stream>>>(dinv, N);

    // 2) h1 = x @ W1  (WMMA fp32, one wave per 16-row tile, 8 waves/block)
    {
        int tiles  = N / 16;                   // 6250, exact
        int blocks = (tiles + 7) / 8;
        k_gemm1_wmma<<<blocks, B, 0, stream>>>(x, W1, h1, N);
    }

    // 3) layer-1 aggregate
    k_init_acc1<<<((N * FMID) + B - 1) / B, B, 0, stream>>>(h1, dinv, acc1, N);
    {
        long long work = (long long)E * FMID;
        int blocks = (int)((work + B - 1) / B);
        k_scatter1<<<blocks, B, 0, stream>>>(row, col, dinv, h1, acc1, E);
    }

    // 4) relu + b1, dot with W2, seed layer-2 accumulator with self-loop term
    k_relu_dot<<<(N + B - 1) / B, B, 0, stream>>>(acc1, b1, W2, dinv, tnode, out_acc, N);

    // 5) layer-2 aggregate + finalize
    k_scatter2<<<(E + B - 1) / B, B, 0, stream>>>(row, col, dinv, tnode, out_acc, E);
    k_finalize<<<(N + B - 1) / B, B, 0, stream>>>(out_acc, b2, out, N);
}